// GATLayer_40870908789447
// MI455X (gfx1250) — compile-verified
//
#include <hip/hip_runtime.h>
#include <hip/hip_bf16.h>

// Problem constants (from reference): N=6144, E=196608, F_IN=256, H=4, D=64
#define N_NODES 6144
#define F_IN    256
#define HD      256   // H*D
#define NHEAD   4
#define HDIM    64

typedef __attribute__((ext_vector_type(2))) float v2f;
typedef __attribute__((ext_vector_type(8))) float v8f;

// ---------------------------------------------------------------------------
// Init: zero the atomic accumulation buffers (out, denom) and set maxsc=-inf.
// ---------------------------------------------------------------------------
__global__ void gat_init_kernel(float* __restrict__ out,
                                float* __restrict__ maxsc,
                                float* __restrict__ denom) {
    int idx = blockIdx.x * blockDim.x + threadIdx.x;
    if (idx < N_NODES * HD) out[idx] = 0.0f;
    if (idx < N_NODES * NHEAD) {
        maxsc[idx] = -1e30f;
        denom[idx] = 0.0f;
    }
}

// ---------------------------------------------------------------------------
// Phase 1: Wh = X @ W using V_WMMA_F32_16X16X4_F32 (full f32 precision).
// Block = 256 threads = 8 waves. Each wave owns a 16x16 output tile;
// block covers 128 rows x 16 cols. W column-slice staged in LDS.
// Prefetch of the A row is hoisted: one global_prefetch_b8 per 128B line,
// keeping the pipelined inner loop at pure load->wmma.
// ---------------------------------------------------------------------------
__global__ __launch_bounds__(256) void gat_gemm_wh_wmma(
        const float* __restrict__ X,   // [N, F_IN]
        const float* __restrict__ W,   // [F_IN, HD]
        float* __restrict__ Wh) {      // [N, HD]
    __shared__ float Wtile[F_IN * 16];         // [k][n], n fastest — 16 KB

    const int colTile  = blockIdx.x & 15;      // 16 column tiles (HD/16)
    const int blockRow = blockIdx.x >> 4;      // 48 row blocks (N/128)
    const int colBase  = colTile * 16;
    const int tid      = threadIdx.x;

    // Cooperative load of the 256x16 W slice into LDS.
    for (int i = tid; i < F_IN * 16; i += 256) {
        const int k = i >> 4, n = i & 15;
        Wtile[i] = W[k * HD + colBase + n];
    }

    const int wave    = tid >> 5;
    const int lane    = tid & 31;
    const int rowBase = blockRow * 128 + wave * 16;
    const int mrow    = rowBase + (lane & 15); // A-frag: M = lane&15
    const int kk      = (lane >> 4) * 2;       // upper half-lanes hold K+2, K+3
    const int ncol    = lane & 15;             // B/D-frag: N = lane&15

    const float* __restrict__ arow = X + (size_t)mrow * F_IN;

    // Prefetch the full 1 KB A row once (8 x 128B cachelines).
#pragma unroll
    for (int pf = 0; pf < F_IN; pf += 32) {
        __builtin_prefetch(arow + pf, 0, 1);
    }

    __syncthreads();

    v8f acc = {};
    for (int k0 = 0; k0 < F_IN; k0 += 4) {
        // A fragment (16x4 f32): v0 = K, v1 = K+1 (upper half-lanes shifted by 2).
        const float2 av = *(const float2*)(arow + k0 + kk);
        v2f a; a.x = av.x; a.y = av.y;

        // B fragment (4x16 f32): lanes 0-15 hold rows K,K+1; lanes 16-31 rows K+2,K+3.
        v2f b;
        b.x = Wtile[(k0 + kk)     * 16 + ncol];
        b.y = Wtile[(k0 + kk + 1) * 16 + ncol];

        // 8 args: (neg_a, A, neg_b, B, c_mod, C, reuse_a, reuse_b)
        acc = __builtin_amdgcn_wmma_f32_16x16x4_f32(
            false, a, false, b, (short)0, acc, false, false);
    }

    // D layout: VGPR i holds row i (lanes 0-15) / row i+8 (lanes 16-31), col = lane&15.
    const int rOff = 8 * (lane >> 4);
#pragma unroll
    for (int i = 0; i < 8; ++i) {
        Wh[(size_t)(rowBase + rOff + i) * HD + colBase + ncol] = acc[i];
    }
}

// ---------------------------------------------------------------------------
// Phase 2: per-node/head attention scores.
// ---------------------------------------------------------------------------
__global__ void gat_scores_kernel(const float* __restrict__ Wh,
                                  const float* __restrict__ a_src,
                                  const float* __restrict__ a_dst,
                                  float* __restrict__ src_score,
                                  float* __restrict__ dst_score) {
    const int idx = blockIdx.x * blockDim.x + threadIdx.x;
    if (idx >= N_NODES * NHEAD) return;
    const int n = idx >> 2, h = idx & 3;
    const float* __restrict__ w  = Wh + (size_t)n * HD + h * HDIM;
    const float* __restrict__ as = a_src + h * HDIM;
    const float* __restrict__ ad = a_dst + h * HDIM;
    float ss = 0.0f, ds = 0.0f;
#pragma unroll 8
    for (int d = 0; d < HDIM; ++d) {
        const float v = w[d];
        ss += v * as[d];
        ds += v * ad[d];
    }
    src_score[idx] = ss;
    dst_score[idx] = ds;
}

// Order-independent float atomic max via CAS.
__device__ __forceinline__ void atomicMaxF(float* addr, float val) {
    unsigned int* ua = (unsigned int*)addr;
    unsigned int old = *ua;
    while (__uint_as_float(old) < val) {
        const unsigned int assumed = old;
        old = atomicCAS(ua, assumed, __float_as_uint(val));
        if (old == assumed) break;
    }
}

// ---------------------------------------------------------------------------
// Phase 3a: edge scores (leaky-relu) + per-destination running max.
// ---------------------------------------------------------------------------
__global__ void gat_edge_max_kernel(const int* __restrict__ edge_index, int E,
                                    const float* __restrict__ src_score,
                                    const float* __restrict__ dst_score,
                                    float* __restrict__ edge_score,
                                    float* __restrict__ maxsc) {
    const int idx = blockIdx.x * blockDim.x + threadIdx.x;
    if (idx >= E * NHEAD) return;
    const int e = idx >> 2, h = idx & 3;
    const int src = edge_index[e];
    const int dst = edge_index[E + e];
    float s = src_score[src * NHEAD + h] + dst_score[dst * NHEAD + h];
    s = (s > 0.0f) ? s : 0.2f * s;               // leaky_relu, slope 0.2
    edge_score[idx] = s;
    atomicMaxF(&maxsc[dst * NHEAD + h], s);
}

// ---------------------------------------------------------------------------
// Phase 3b: w = exp(s - max[dst]); denom[dst] += w; out[dst] += w * Wh[src].
// All accumulators are L2-resident (~10 MB), so atomics stay on-chip.
// ---------------------------------------------------------------------------
__global__ void gat_edge_accum_kernel(const int* __restrict__ edge_index, int E,
                                      const float* __restrict__ edge_score,
                                      const float* __restrict__ maxsc,
                                      const float* __restrict__ Wh,
                                      float* __restrict__ denom,
                                      float* __restrict__ out) {
    const int idx = blockIdx.x * blockDim.x + threadIdx.x;
    if (idx >= E * NHEAD) return;
    const int e = idx >> 2, h = idx & 3;
    const int src = edge_index[e];
    const int dst = edge_index[E + e];
    const float m = maxsc[dst * NHEAD + h];
    const float w = __expf(edge_score[idx] - m);
    atomicAdd(&denom[dst * NHEAD + h], w);
    const float* __restrict__ ws = Wh + (size_t)src * HD + h * HDIM;
    float* __restrict__ o = out + (size_t)dst * HD + h * HDIM;
#pragma unroll 8
    for (int d = 0; d < HDIM; ++d) {
        atomicAdd(&o[d], w * ws[d]);
    }
}

// ---------------------------------------------------------------------------
// Phase 4: normalize by the softmax denominator.
// ---------------------------------------------------------------------------
__global__ void gat_normalize_kernel(const float* __restrict__ denom,
                                     float* __restrict__ out) {
    const int idx = blockIdx.x * blockDim.x + threadIdx.x;
    if (idx >= N_NODES * HD) return;
    const int n = idx >> 8;            // /HD
    const int h = (idx >> 6) & 3;      // head of this column
    const float dn = denom[n * NHEAD + h];
    const float inv = (dn > 0.0f) ? (1.0f / dn) : 0.0f;
    out[idx] *= inv;
}

// ---------------------------------------------------------------------------
extern "C" void kernel_launch(void* const* d_in, const int* in_sizes, int n_in,
                              void* d_out, int out_size, void* d_ws, size_t ws_size,
                              hipStream_t stream) {
    const float* X     = (const float*)d_in[0];   // [N, F_IN]
    const int*   eidx  = (const int*)d_in[1];     // [2, E]
    const float* W     = (const float*)d_in[2];   // [F_IN, HD]
    const float* a_src = (const float*)d_in[3];   // [H, D]
    const float* a_dst = (const float*)d_in[4];   // [H, D]
    float* out = (float*)d_out;                   // [N, HD]

    const int E = in_sizes[1] / 2;

    // Workspace carve-up (all f32).
    float* ws        = (float*)d_ws;
    float* Wh        = ws;                                      // N*HD
    float* src_score = Wh + (size_t)N_NODES * HD;               // N*H
    float* dst_score = src_score + N_NODES * NHEAD;             // N*H
    float* maxsc     = dst_score + N_NODES * NHEAD;             // N*H
    float* denom     = maxsc + N_NODES * NHEAD;                 // N*H
    float* escore    = denom + N_NODES * NHEAD;                 // E*H

    const int threads = 256;

    // 1) init accumulators
    gat_init_kernel<<<(N_NODES * HD + threads - 1) / threads, threads, 0, stream>>>(
        out, maxsc, denom);

    // 2) Wh = X @ W  (WMMA f32 16x16x4)
    gat_gemm_wh_wmma<<<(N_NODES / 128) * (HD / 16), threads, 0, stream>>>(X, W, Wh);

    // 3) per-node scores
    gat_scores_kernel<<<(N_NODES * NHEAD + threads - 1) / threads, threads, 0, stream>>>(
        Wh, a_src, a_dst, src_score, dst_score);

    // 4) edge scores + per-dst max
    const int egrid = (E * NHEAD + threads - 1) / threads;
    gat_edge_max_kernel<<<egrid, threads, 0, stream>>>(
        eidx, E, src_score, dst_score, escore, maxsc);

    // 5) exp-weighted accumulate
    gat_edge_accum_kernel<<<egrid, threads, 0, stream>>>(
        eidx, E, escore, maxsc, Wh, denom, out);

    // 6) normalize
    gat_normalize_kernel<<<(N_NODES * HD + threads - 1) / threads, threads, 0, stream>>>(
        denom, out);
}